// RelAttXL_11201274708212
// MI455X (gfx1250) — compile-verified
//
#include <hip/hip_runtime.h>

// ---------------- CDNA5 WMMA f32 16x16x4 ----------------
typedef __attribute__((ext_vector_type(2))) float v2f;
typedef __attribute__((ext_vector_type(8))) float v8f;

__device__ __forceinline__ v8f wmma4(v2f a, v2f b, v8f c) {
  // 8 args: (neg_a, A, neg_b, B, c_mod, C, reuse_a, reuse_b)
  return __builtin_amdgcn_wmma_f32_16x16x4_f32(false, a, false, b, (short)0, c,
                                               false, false);
}

#define NB     4
#define NSEQ   1024
#define DMODEL 1024
#define NHEAD  16
#define DHEAD  64
#define NPOS   2047  // 2n-1

// ---------------- Tiled GEMM: C[M,N] = A[M,K] @ B[K,N] (+ bias row) ---------
// block = 256 threads = 8 waves; block tile 64(M) x 128(N); K chunk 32.
// A staged row-major [64][36]; B staged TRANSPOSED [col][k] so each lane's
// (B[k][col], B[k+1][col]) pair is one contiguous ds_load_b64.
__global__ __launch_bounds__(256) void gemm_bias(
    const float* __restrict__ A, const float* __restrict__ B,
    const float* __restrict__ bias, float* __restrict__ C,
    int M, int N, int K)
{
  __shared__ float As[64][36];     // pad 36: 16 consecutive rows -> 16 banks
  __shared__ float Bst[128][34];   // pad 34 (even): conflict-free, 8B-aligned

  const int tid  = threadIdx.x;
  const int lane = tid & 31;
  const int wave = tid >> 5;
  const int r    = lane & 15;
  const int half = lane >> 4;
  const int koff = half * 2;
  const int wM = (wave >> 1) * 16;   // 0,16,32,48
  const int wN = (wave & 1) * 64;    // 0,64
  const int m0 = blockIdx.y * 64;
  const int n0 = blockIdx.x * 128;

  v8f acc[4] = {};

  for (int k0 = 0; k0 < K; k0 += 32) {
    // stage A: 64 rows x 32 cols (512 float4, 2 per thread)
#pragma unroll
    for (int i = 0; i < 2; ++i) {
      int lin = tid + i * 256;
      int row = lin >> 3;
      int c4  = (lin & 7) * 4;
      int gm  = m0 + row; if (gm >= M) gm = M - 1;
      *(float4*)&As[row][c4] =
          *(const float4*)&A[(size_t)gm * K + k0 + c4];
    }
    // stage B transposed: read float4 along N, scatter into Bst[col][k]
#pragma unroll
    for (int i = 0; i < 4; ++i) {
      int lin = tid + i * 256;
      int row = lin >> 5;          // k within chunk: 0..31
      int c4  = (lin & 31) * 4;    // col: 0..124
      float4 t4 = *(const float4*)&B[(size_t)(k0 + row) * N + n0 + c4];
      Bst[c4 + 0][row] = t4.x;
      Bst[c4 + 1][row] = t4.y;
      Bst[c4 + 2][row] = t4.z;
      Bst[c4 + 3][row] = t4.w;
    }
    __syncthreads();

#pragma unroll
    for (int kk = 0; kk < 32; kk += 4) {
      v2f a = *(const v2f*)&As[wM + r][kk + koff];
#pragma unroll
      for (int t = 0; t < 4; ++t) {
        v2f bb = *(const v2f*)&Bst[wN + t * 16 + r][kk + koff];
        acc[t] = wmma4(a, bb, acc[t]);
      }
    }
    __syncthreads();
  }

#pragma unroll
  for (int t = 0; t < 4; ++t) {
    int gcol  = n0 + wN + t * 16 + r;
    float bv = bias ? bias[gcol] : 0.0f;
#pragma unroll
    for (int i = 0; i < 8; ++i) {
      int grow = m0 + wM + i + 8 * half;   // C/D layout: M = i + 8*half
      if (grow < M) C[(size_t)grow * N + gcol] = acc[t][i] + bv;
    }
  }
}

// ---------------- Scores: out[bh,i,j] = (q[b,i,h,:]+hb[h,:]) . kp[b,j,h,:] --
// One 16x16 tile per wave, K = DHEAD = 64. kp has Ncols rows per batch.
__global__ __launch_bounds__(128) void score_gemm(
    const float* __restrict__ q, const float* __restrict__ kp,
    const float* __restrict__ hbias, float* __restrict__ out, int Ncols)
{
  const int lane = threadIdx.x & 31;
  const int wave = threadIdx.x >> 5;
  const int r = lane & 15, half = lane >> 4, koff = half * 2;
  const int bh = blockIdx.z;
  const int b  = bh >> 4, h = bh & 15;
  const int i0 = blockIdx.y * 16;
  const int j0 = (blockIdx.x * 4 + wave) * 16;

  const int j  = j0 + r;
  const int jc = (j < Ncols) ? j : (Ncols - 1);

  const float* qrow = q  + ((size_t)(b * NSEQ + i0 + r)) * DMODEL + h * DHEAD;
  const float* hb   = hbias + h * DHEAD;
  const float* krow = kp + ((size_t)b * Ncols + jc) * DMODEL + h * DHEAD;

  v8f acc = {};
#pragma unroll
  for (int kd = 0; kd < DHEAD; kd += 4) {
    v2f a  = *(const v2f*)&qrow[kd + koff];
    v2f ub = *(const v2f*)&hb[kd + koff];
    v2f bb = *(const v2f*)&krow[kd + koff];   // B[kd][j] = kp[j][kd]
    a = a + ub;
    acc = wmma4(a, bb, acc);
  }

  if (j < Ncols) {
#pragma unroll
    for (int i = 0; i < 8; ++i) {
      out[((size_t)bh * NSEQ + (i0 + i + 8 * half)) * Ncols + j] = acc[i];
    }
  }
}

// ---------------- Softmax over shifted scores -------------------------------
// shifted pos: pos[bh,i, j + n-1-i]  (contiguous in j). 1 block per row.
__global__ __launch_bounds__(256) void softmax_kernel(
    const float* __restrict__ cont, const float* __restrict__ pos,
    float* __restrict__ attn)
{
  const int row = blockIdx.x;            // bh*1024 + i
  const int bh  = row >> 10;
  const int i   = row & 1023;
  const int tid = threadIdx.x;
  const float scale = 0.03125f;          // 1/sqrt(1024)

  const float* c = cont + (size_t)row * NSEQ;
  const float* p = pos  + (size_t)bh * NSEQ * NPOS + (size_t)i * NPOS
                        + (NSEQ - 1 - i);
  __shared__ float red[256];
  float e[4];
  float m = -1e30f;
#pragma unroll
  for (int t = 0; t < 4; ++t) {
    int jj = tid + t * 256;
    e[t] = (c[jj] + p[jj]) * scale;
    m = fmaxf(m, e[t]);
  }
  red[tid] = m; __syncthreads();
  for (int s = 128; s > 0; s >>= 1) {
    if (tid < s) red[tid] = fmaxf(red[tid], red[tid + s]);
    __syncthreads();
  }
  m = red[0]; __syncthreads();

  float sum = 0.0f;
#pragma unroll
  for (int t = 0; t < 4; ++t) { e[t] = __expf(e[t] - m); sum += e[t]; }
  red[tid] = sum; __syncthreads();
  for (int s = 128; s > 0; s >>= 1) {
    if (tid < s) red[tid] += red[tid + s];
    __syncthreads();
  }
  const float inv = 1.0f / red[0];
  float* arow = attn + (size_t)row * NSEQ;
#pragma unroll
  for (int t = 0; t < 4; ++t) arow[tid + t * 256] = e[t] * inv;
}

// ---------------- ctx[b,i,h,:] = attn[bh,i,:] @ v[b,:,h,:] ------------------
// 4 waves per block cover the 64 head-dim cols; 1 row-tile per block.
__global__ __launch_bounds__(128) void attnv_kernel(
    const float* __restrict__ attn, const float* __restrict__ vmat,
    float* __restrict__ ctx)
{
  const int lane = threadIdx.x & 31;
  const int wave = threadIdx.x >> 5;     // col tile 0..3
  const int r = lane & 15, half = lane >> 4, koff = half * 2;
  const int bh = blockIdx.y;
  const int b  = bh >> 4, h = bh & 15;
  const int i0 = blockIdx.x * 16;
  const int col = wave * 16 + r;

  const float* arow = attn + ((size_t)bh * NSEQ + i0 + r) * NSEQ;
  const float* vcol = vmat + (size_t)b * NSEQ * DMODEL + h * DHEAD + col;

  v8f acc = {};
  for (int m = 0; m < NSEQ; m += 4) {
    v2f a = *(const v2f*)&arow[m + koff];
    v2f bb;
    bb.x = vcol[(size_t)(m + koff)     * DMODEL];
    bb.y = vcol[(size_t)(m + koff + 1) * DMODEL];
    acc  = wmma4(a, bb, acc);
  }
#pragma unroll
  for (int i = 0; i < 8; ++i) {
    ctx[((size_t)b * NSEQ + i0 + i + 8 * half) * DMODEL + h * DHEAD + col] =
        acc[i];
  }
}

// ---------------- Launcher ---------------------------------------------------
extern "C" void kernel_launch(void* const* d_in, const int* in_sizes, int n_in,
                              void* d_out, int out_size, void* d_ws,
                              size_t ws_size, hipStream_t stream)
{
  const float* query   = (const float*)d_in[0];
  const float* key_    = (const float*)d_in[1];
  const float* value   = (const float*)d_in[2];
  const float* pos_emb = (const float*)d_in[3];
  const float* Wq = (const float*)d_in[4];
  const float* bq = (const float*)d_in[5];
  const float* Wk = (const float*)d_in[6];
  const float* bk = (const float*)d_in[7];
  const float* Wv = (const float*)d_in[8];
  const float* bv = (const float*)d_in[9];
  const float* Wo = (const float*)d_in[10];
  const float* bo = (const float*)d_in[11];
  const float* Wp = (const float*)d_in[12];
  const float* u_bias = (const float*)d_in[13];
  const float* v_bias = (const float*)d_in[14];

  float* out = (float*)d_out;
  float* ws  = (float*)d_ws;

  const size_t SZ_QKV = (size_t)NB * NSEQ * DMODEL;   // 4,194,304 floats
  const size_t SZ_P   = (size_t)NB * NPOS * DMODEL;   // 8,384,512 floats
  const size_t SZ_SC  = (size_t)NB * NHEAD * NSEQ * NSEQ;

  float* q   = ws;                 // projected q
  float* k   = q   + SZ_QKV;       // projected k
  float* vv  = k   + SZ_QKV;       // projected v
  float* p   = vv  + SZ_QKV;       // projected pos
  float* ctx = p   + SZ_P;         // attn @ v

  float* o_out  = out;                     // [b,n,d]
  float* o_attn = o_out  + SZ_QKV;         // [b,h,n,n]
  float* o_cont = o_attn + SZ_SC;          // [b,h,n,n]
  float* o_pos  = o_cont + SZ_SC;          // [b,h,n,2n-1]

  const dim3 gblk(256);
  const dim3 gqkv(DMODEL / 128, (NB * NSEQ) / 64);          // (8, 64)
  const dim3 gp(DMODEL / 128, (NB * NPOS + 63) / 64);       // (8, 128)

  // Projections
  gemm_bias<<<gqkv, gblk, 0, stream>>>(query,   Wq, bq,      q,  NB * NSEQ, DMODEL, DMODEL);
  gemm_bias<<<gqkv, gblk, 0, stream>>>(key_,    Wk, bk,      k,  NB * NSEQ, DMODEL, DMODEL);
  gemm_bias<<<gqkv, gblk, 0, stream>>>(value,   Wv, bv,      vv, NB * NSEQ, DMODEL, DMODEL);
  gemm_bias<<<gp,   gblk, 0, stream>>>(pos_emb, Wp, nullptr, p,  NB * NPOS, DMODEL, DMODEL);

  // Scores (content: N=1024, pos: N=2047)
  const dim3 sblk(128);
  const dim3 gcont(NSEQ / 64,        NSEQ / 16, NB * NHEAD); // (16, 64, 64)
  const dim3 gpos((NPOS + 63) / 64,  NSEQ / 16, NB * NHEAD); // (32, 64, 64)
  score_gemm<<<gcont, sblk, 0, stream>>>(q, k, u_bias, o_cont, NSEQ);
  score_gemm<<<gpos,  sblk, 0, stream>>>(q, p, v_bias, o_pos,  NPOS);

  // Shifted softmax -> attn
  softmax_kernel<<<dim3(NB * NHEAD * NSEQ), dim3(256), 0, stream>>>(
      o_cont, o_pos, o_attn);

  // attn @ v -> ctx, then output projection
  attnv_kernel<<<dim3(NSEQ / 16, NB * NHEAD), sblk, 0, stream>>>(o_attn, vv, ctx);
  gemm_bias<<<gqkv, gblk, 0, stream>>>(ctx, Wo, bo, o_out, NB * NSEQ, DMODEL, DMODEL);
}